// NnaQueue_48670569398428
// MI455X (gfx1250) — compile-verified
//
#include <hip/hip_runtime.h>

#define BATCH   2048
#define DIM     64
#define QSIZE   131072
#define HALFB   1024          // rows enqueued = BATCH/2

typedef __attribute__((ext_vector_type(2))) float v2f;
typedef __attribute__((ext_vector_type(8))) float v8f;

// ---- packed argmax key: [63:32] order-preserving float bits, [31:0] = ~col
// max(key) == (max value, then lowest column index) -> matches jnp.argmax ties.
__device__ __forceinline__ unsigned long long packKey(float v, unsigned int col) {
    unsigned int u = __float_as_uint(v);
    u = (u & 0x80000000u) ? ~u : (u | 0x80000000u);   // monotonic total order
    return ((unsigned long long)u << 32) | (unsigned long long)(0xFFFFFFFFu ^ col);
}

__device__ __forceinline__ unsigned long long shflXor64(unsigned long long v, int mask) {
    unsigned int lo = (unsigned int)v;
    unsigned int hi = (unsigned int)(v >> 32);
    lo = __shfl_xor(lo, mask, 32);
    hi = __shfl_xor(hi, mask, 32);
    return ((unsigned long long)hi << 32) | (unsigned long long)lo;
}

// ---------------- kernel 1: clear per-row best keys ----------------
__global__ void nnq_init_best(unsigned long long* best) {
    int i = blockIdx.x * blockDim.x + threadIdx.x;
    if (i < BATCH) best[i] = 0ull;
}

// ---------------- kernel 2: WMMA similarity + running argmax ----------------
// grid: (128 batch tiles, 16 col groups), block: 256 (8 waves).
// Each wave: 16 rows x 1024 cols strip = 64 tiles x 16 chained f32 WMMAs.
__global__ __launch_bounds__(256)
void nnq_sim_argmax(const float* __restrict__ x,
                    const float* __restrict__ queue,
                    unsigned long long* __restrict__ best) {
    const int lane   = threadIdx.x & 31;
    const int wave   = threadIdx.x >> 5;
    const int m0     = blockIdx.x * 16;
    const int chunk  = blockIdx.y * 8 + wave;   // 0..127
    const int nstart = chunk << 10;             // 1024 cols per wave
    const int mrow   = lane & 15;               // row within tile (A) / col within tile (B)
    const int khalf  = lane >> 4;               // K pair select (K=0,1 vs K=2,3)

    // A fragments: x tile 16x64, resident for whole strip.
    v2f a[16];
#pragma unroll
    for (int s = 0; s < 16; ++s)
        a[s] = *(const v2f*)(x + (m0 + mrow) * DIM + 4 * s + 2 * khalf);

    unsigned long long bk[8];
#pragma unroll
    for (int r = 0; r < 8; ++r) bk[r] = 0ull;

    for (int t = 0; t < 64; ++t) {
        const int n0 = nstart + t * 16;
        const float* qb = queue + (size_t)(n0 + mrow) * DIM + 2 * khalf;
        v8f c = {0.f, 0.f, 0.f, 0.f, 0.f, 0.f, 0.f, 0.f};
#pragma unroll
        for (int s = 0; s < 16; ++s) {
            v2f b = *(const v2f*)(qb + 4 * s);
            c = __builtin_amdgcn_wmma_f32_16x16x4_f32(
                    /*neg_a=*/false, a[s], /*neg_b=*/false, b,
                    /*c_mod=*/(short)0, c, /*reuse_a=*/false, /*reuse_b=*/false);
        }
        // C/D layout: VGPR r, lanes 0-15 -> M=r, lanes 16-31 -> M=r+8; N = lane&15
        const unsigned int col = (unsigned int)(n0 + mrow);
#pragma unroll
        for (int r = 0; r < 8; ++r) {
            unsigned long long k = packKey(c[r], col);
            bk[r] = (k > bk[r]) ? k : bk[r];
        }
    }

    // reduce across the 16 lanes of each half (halves hold disjoint rows)
#pragma unroll
    for (int r = 0; r < 8; ++r) {
        unsigned long long k = bk[r];
#pragma unroll
        for (int m = 1; m <= 8; m <<= 1) {
            unsigned long long o = shflXor64(k, m);
            k = (o > k) ? o : k;
        }
        if (mrow == 0)
            atomicMax(&best[m0 + r + 8 * khalf], k);
    }
}

// ---------------- kernel 3: gather nearest-neighbor rows ----------------
__global__ void nnq_gather(const unsigned long long* __restrict__ best,
                           const float* __restrict__ queue,
                           float* __restrict__ nn) {
    const int row = blockIdx.x;                 // one block per batch row, 64 threads
    const unsigned long long k = best[row];
    const unsigned int idx = 0xFFFFFFFFu ^ (unsigned int)k;
    nn[(size_t)row * DIM + threadIdx.x] = queue[(size_t)idx * DIM + threadIdx.x];
}

// ---------------- kernel 4: queue copy + enqueue window + new ptr ----------------
__global__ __launch_bounds__(256)
void nnq_update(const float* __restrict__ x,
                const float* __restrict__ queue,
                const int* __restrict__ ptr_p,
                float* __restrict__ out_queue,
                float* __restrict__ out_ptr) {
    const int gid = blockIdx.x * blockDim.x + threadIdx.x;  // float4 index
    const int ptr = *ptr_p;
    const int row = gid >> 4;                                // 16 float4 per row
    const int c4  = gid & 15;
    const int off = (row - ptr) & (QSIZE - 1);               // wraparound distance
    float4 v;
    if (off < HALFB)
        v = ((const float4*)x)[off * 16 + c4];
    else
        v = ((const float4*)queue)[gid];
    ((float4*)out_queue)[gid] = v;
    if (gid == 0)
        *out_ptr = (float)((ptr + HALFB) & (QSIZE - 1));
}

extern "C" void kernel_launch(void* const* d_in, const int* in_sizes, int n_in,
                              void* d_out, int out_size, void* d_ws, size_t ws_size,
                              hipStream_t stream) {
    const float* x     = (const float*)d_in[0];   // [2048,64] f32
    const float* queue = (const float*)d_in[1];   // [131072,64] f32
    const int*   ptr   = (const int*)d_in[2];     // scalar

    float* nn   = (float*)d_out;                          // [2048,64]
    float* outq = nn + (size_t)BATCH * DIM;               // [131072,64]
    float* outp = outq + (size_t)QSIZE * DIM;             // scalar

    unsigned long long* best = (unsigned long long*)d_ws; // 2048 * u64 = 16 KB

    nnq_init_best<<<(BATCH + 255) / 256, 256, 0, stream>>>(best);
    nnq_sim_argmax<<<dim3(BATCH / 16, 16), 256, 0, stream>>>(x, queue, best);
    nnq_gather<<<BATCH, DIM, 0, stream>>>(best, queue, nn);
    nnq_update<<<(QSIZE * DIM / 4) / 256, 256, 0, stream>>>(x, queue, ptr, outq, outp);
}